// Decoder_11295763988568
// MI455X (gfx1250) — compile-verified
//
#include <hip/hip_runtime.h>
#include <math.h>
#include <stdint.h>

typedef float v2f __attribute__((ext_vector_type(2)));
typedef float v8f __attribute__((ext_vector_type(8)));
typedef int   v4i __attribute__((ext_vector_type(4)));

typedef __attribute__((address_space(1))) v4i as1_v4i;
typedef __attribute__((address_space(3))) v4i as3_v4i;

#define H  512
#define B  128
#define S  512
#define BS (B * S)
#define G4H (4 * H)

__device__ __forceinline__ float sigm(float x) { return 1.0f / (1.0f + __expf(-x)); }

__device__ __forceinline__ v2f ld2(const float* p) {
    float2 t = *(const float2*)p;
    v2f r; r.x = t.x; r.y = t.y; return r;
}

// ---- async global->LDS 16-byte copy (GLOBAL_LOAD_ASYNC_TO_LDS_B128) ------
__device__ __forceinline__ void async_copy16(const float* gp, float* lp) {
#if __has_builtin(__builtin_amdgcn_global_load_async_to_lds_b128)
    __builtin_amdgcn_global_load_async_to_lds_b128(
        (as1_v4i*)(uintptr_t)gp,
        (as3_v4i*)(uint32_t)(uintptr_t)lp,   // low 32 bits of generic ptr = LDS offset
        0, 0);
#else
    unsigned ldsoff = (unsigned)(uintptr_t)lp;
    asm volatile("global_load_async_to_lds_b128 %0, %1, off"
                 :: "v"(ldsoff), "v"(gp) : "memory");
#endif
}

#if __has_builtin(__builtin_amdgcn_s_wait_asynccnt)
#define WAIT_ASYNC(n) __builtin_amdgcn_s_wait_asynccnt(n)
#else
#define WAIT_ASYNC(n) asm volatile("s_wait_asynccnt " #n ::: "memory")
#endif

// ---------------- zero the atomic accumulator -----------------------------
__global__ void k_zero(float* p, int n) {
    int i = blockIdx.x * blockDim.x + threadIdx.x;
    if (i < n) p[i] = 0.0f;
}

// ---------------- LSTM gates: gates[128][2048] = x@Wih^T + h@Whh^T --------
__global__ __launch_bounds__(256) void k_lstm_gates(
    const float* __restrict__ x, const float* __restrict__ hprev,
    const float* __restrict__ Wih, const float* __restrict__ Whh,
    const float* __restrict__ bih, const float* __restrict__ bhh,
    float* __restrict__ gates)
{
    int tid = threadIdx.x, wave = tid >> 5, lane = tid & 31;
    int tile = blockIdx.x * 8 + wave;      // 1024 tiles = 8 m-tiles x 128 n-tiles
    int mt = tile >> 7, nt = tile & 127;
    int m0 = mt * 16, n0 = nt * 16;
    int l = lane & 15, halfk = (lane >> 4) * 2;

    v8f acc = {};
    const float* ap[2] = {x, hprev};
    const float* bp[2] = {Wih, Whh};
    for (int p = 0; p < 2; ++p) {
        const float* A  = ap[p];
        const float* Bm = bp[p];
        for (int k = 0; k < H; k += 4) {
            int kb = k + halfk;
            v2f a = ld2(A  + (size_t)(m0 + l) * H + kb);
            v2f b = ld2(Bm + (size_t)(n0 + l) * H + kb);
            acc = __builtin_amdgcn_wmma_f32_16x16x4_f32(false, a, false, b,
                                                        (short)0, acc, false, false);
        }
    }
    float bias = bih[n0 + l] + bhh[n0 + l];
    int rowadd = (lane >> 4) * 8;
    for (int r = 0; r < 8; ++r)
        gates[(size_t)(m0 + r + rowadd) * G4H + n0 + l] = acc[r] + bias;
}

// ---------------- LSTM pointwise ------------------------------------------
__global__ void k_lstm_pointwise(const float* __restrict__ gates,
                                 const float* __restrict__ c,
                                 float* __restrict__ hout, float* __restrict__ cout)
{
    int i = blockIdx.x * blockDim.x + threadIdx.x;   // 65536
    int b = i >> 9, j = i & 511;
    const float* g = gates + (size_t)b * G4H;
    float ig = sigm(g[j]);
    float fg = sigm(g[j + 512]);
    float gg = tanhf(g[j + 1024]);
    float og = sigm(g[j + 1536]);
    float cn = fg * c[i] + ig * gg;
    hout[i] = og * tanhf(cn);
    cout[i] = cn;
}

// ---------------- w2h = h_new @ W2^T + b2  [128][512] ---------------------
__global__ __launch_bounds__(256) void k_w2h(
    const float* __restrict__ hn, const float* __restrict__ W2,
    const float* __restrict__ b2, float* __restrict__ w2h)
{
    int tid = threadIdx.x, wave = tid >> 5, lane = tid & 31;
    int tile = blockIdx.x * 8 + wave;      // 256 tiles = 8 m x 32 n
    int mt = tile >> 5, nt = tile & 31;
    int m0 = mt * 16, n0 = nt * 16;
    int l = lane & 15, halfk = (lane >> 4) * 2;

    v8f acc = {};
    for (int k = 0; k < H; k += 4) {
        int kb = k + halfk;
        v2f a = ld2(hn + (size_t)(m0 + l) * H + kb);
        v2f b = ld2(W2 + (size_t)(n0 + l) * H + kb);
        acc = __builtin_amdgcn_wmma_f32_16x16x4_f32(false, a, false, b,
                                                    (short)0, acc, false, false);
    }
    float bias = b2[n0 + l];
    int rowadd = (lane >> 4) * 8;
    for (int r = 0; r < 8; ++r)
        w2h[(size_t)(m0 + r + rowadd) * H + n0 + l] = acc[r] + bias;
}

// ---------------- fused attention: a[bs] += sum_o tanh(W1*enc^T + w2h)*V --
// D[o, bs] = W1 x enc^T. WG tile: 64 o x 128 bs. K staged through double-
// buffered row-major LDS panels (pitch 20 floats => conflict-free b64 frags)
// filled by GLOBAL_LOAD_ASYNC_TO_LDS_B128 (ASYNCcnt), overlapped with WMMA.
#define PITCH 20
__global__ __launch_bounds__(256, 2) void k_attn(
    const float* __restrict__ enc, const float* __restrict__ W1,
    const float* __restrict__ w2h, const float* __restrict__ V,
    float* __restrict__ a_out)
{
    __shared__ float A_s[2][64][PITCH];    // [buf][o][k]   W1 panel
    __shared__ float B_s[2][128][PITCH];   // [buf][bs][k]  enc panel
    __shared__ float w2h_s[64], V_s[64];

    int o_base  = blockIdx.x * 64;
    int bs_base = blockIdx.y * 128;
    int b = bs_base >> 9;                       // 128 bs-cols lie inside one batch
    int t = threadIdx.x, wave = t >> 5, lane = t & 31;
    int wo = wave >> 2, wn = wave & 3;
    int l = lane & 15, hsel = lane >> 4;

    if (t < 64) { w2h_s[t] = w2h[(size_t)b * H + o_base + t]; V_s[t] = V[o_base + t]; }

    // staging maps: A: 4 lanes/row (64B rows), B: 2 lanes/row, 2 copies each
    int rA = t >> 2, kA = (t & 3) * 4;
    int rB = t >> 1, kB = (t & 1) * 8;
    const float* gA = W1  + (size_t)(o_base  + rA) * H + kA;
    const float* gB = enc + (size_t)(bs_base + rB) * H + kB;

#define STAGE(buf, kc)                                         \
    do {                                                       \
        async_copy16(gA + (kc),     &A_s[buf][rA][kA]);        \
        async_copy16(gB + (kc),     &B_s[buf][rB][kB]);        \
        async_copy16(gB + (kc) + 4, &B_s[buf][rB][kB + 4]);    \
    } while (0)

    v8f acc[2][2] = {};
    STAGE(0, 0);
    for (int kc = 0; kc < H; kc += 16) {
        int buf = (kc >> 4) & 1;
        if (kc + 16 < H) {
            STAGE(buf ^ 1, kc + 16);
            WAIT_ASYNC(3);                      // oldest 3 = current panel done
        } else {
            WAIT_ASYNC(0);
        }
        __syncthreads();

        for (int ks = 0; ks < 16; ks += 4) {
            int kb = ks + 2 * hsel;             // lanes<16: K k,k+1 ; lanes>=16: k+2,k+3
            v2f a0 = ld2(&A_s[buf][wo * 32 + l][kb]);
            v2f a1 = ld2(&A_s[buf][wo * 32 + 16 + l][kb]);
            v2f b0 = ld2(&B_s[buf][wn * 32 + l][kb]);
            v2f b1 = ld2(&B_s[buf][wn * 32 + 16 + l][kb]);
            acc[0][0] = __builtin_amdgcn_wmma_f32_16x16x4_f32(false, a0, false, b0, (short)0, acc[0][0], false, false);
            acc[0][1] = __builtin_amdgcn_wmma_f32_16x16x4_f32(false, a0, false, b1, (short)0, acc[0][1], false, false);
            acc[1][0] = __builtin_amdgcn_wmma_f32_16x16x4_f32(false, a1, false, b0, (short)0, acc[1][0], false, false);
            acc[1][1] = __builtin_amdgcn_wmma_f32_16x16x4_f32(false, a1, false, b1, (short)0, acc[1][1], false, false);
        }
        __syncthreads();                        // all waves done with buf before restage
    }

    // Epilogue: u = tanh(D + w2h[o]); partial a[bs] = sum_o u * V[o].
    // C layout: VGPR r = row (r | r+8), lane%16 = column.
    for (int j = 0; j < 2; ++j) {
        float s = 0.0f;
        for (int i = 0; i < 2; ++i) {
            for (int r = 0; r < 8; ++r) {
                int orow = wo * 32 + i * 16 + r + hsel * 8;
                float u = tanhf(acc[i][j][r] + w2h_s[orow]);
                s += u * V_s[orow];
            }
        }
        s += __shfl_xor(s, 16, 32);             // combine row halves (lanes L, L+16)
        if (hsel == 0)
            atomicAdd(&a_out[bs_base + wn * 32 + j * 16 + l], s);
    }
}

// ---------------- masked softmax over S with 10*tanh scaling --------------
__global__ void k_softmax(const float* __restrict__ a, const float* __restrict__ mask,
                          float* __restrict__ policy)
{
    __shared__ float red[256];
    int b = blockIdx.x, t = threadIdx.x;
    float v0 = 10.0f * tanhf(a[b * S + t])       + logf(mask[b * S + t]);
    float v1 = 10.0f * tanhf(a[b * S + t + 256]) + logf(mask[b * S + t + 256]);
    red[t] = fmaxf(v0, v1);
    __syncthreads();
    for (int off = 128; off; off >>= 1) { if (t < off) red[t] = fmaxf(red[t], red[t + off]); __syncthreads(); }
    float m = red[0];
    __syncthreads();
    float e0 = __expf(v0 - m), e1 = __expf(v1 - m);
    red[t] = e0 + e1;
    __syncthreads();
    for (int off = 128; off; off >>= 1) { if (t < off) red[t] += red[t + off]; __syncthreads(); }
    float inv = 1.0f / red[0];
    policy[b * S + t]       = e0 * inv;
    policy[b * S + t + 256] = e1 * inv;
}

extern "C" void kernel_launch(void* const* d_in, const int* in_sizes, int n_in,
                              void* d_out, int out_size, void* d_ws, size_t ws_size,
                              hipStream_t stream)
{
    (void)in_sizes; (void)n_in; (void)out_size; (void)ws_size;
    const float* x    = (const float*)d_in[0];
    const float* h    = (const float*)d_in[1];
    const float* c    = (const float*)d_in[2];
    const float* enc  = (const float*)d_in[3];
    const float* mask = (const float*)d_in[4];
    const float* W1   = (const float*)d_in[5];
    const float* W2   = (const float*)d_in[6];
    const float* b2   = (const float*)d_in[7];
    const float* V    = (const float*)d_in[8];
    const float* Wih  = (const float*)d_in[9];
    const float* Whh  = (const float*)d_in[10];
    const float* bih  = (const float*)d_in[11];
    const float* bhh  = (const float*)d_in[12];

    float* out    = (float*)d_out;
    float* policy = out;                 // [B,S]
    float* hout   = out + BS;            // [B,H]
    float* cout   = out + 2 * BS;        // [B,H]

    float* ws      = (float*)d_ws;
    float* gates_w = ws;                 // 128*2048
    float* w2h_w   = ws + B * G4H;       // 128*512
    float* a_w     = w2h_w + B * H;      // 128*512 (atomic accumulator)

    k_zero<<<BS / 256, 256, 0, stream>>>(a_w, BS);
    k_lstm_gates<<<128, 256, 0, stream>>>(x, h, Wih, Whh, bih, bhh, gates_w);
    k_lstm_pointwise<<<(B * H) / 256, 256, 0, stream>>>(gates_w, c, hout, cout);
    k_w2h<<<32, 256, 0, stream>>>(hout, W2, b2, w2h_w);
    k_attn<<<dim3(8, 512), 256, 0, stream>>>(enc, W1, w2h_w, V, a_w);
    k_softmax<<<B, 256, 0, stream>>>(a_w, mask, policy);
}